// xLSTMBlock_43301860278404
// MI455X (gfx1250) — compile-verified
//
#include <hip/hip_runtime.h>
#include <math.h>

// xLSTM (mLSTM matrix-memory) block for MI455X / gfx1250.
// L=2, H=512, IN=512, B=4, T=128.
//
//  * C state ([B,H,H] f32 = 4MB/layer) resident in LDS: 32 WGs, each owns a
//    (batch, 64-col) slice = 128KB of 320KB/WGP -> no HBM traffic for the
//    recurrent state (would otherwise be ~2GB over T).
//  * fp32 GEMMs via V_WMMA_F32_16X16X4_F32 (native CDNA5 fp32 matrix op).
//  * A-operands (x_t, h_prev; 4x512 each) staged to LDS per step via
//    global_load_async_to_lds_b128 (+ s_wait_asynccnt) when available.
//  * Persistent kernel + global atomic barrier (2 syncs per timestep).

#define L_LAYERS 2
#define H 512
#define IN_DIM 512
#define B_BATCH 4
#define T_STEPS 128
#define THREEH 1536
#define GCOLS 3072          // [i f o | q k v] columns per batch row
#define NWG 32
#define TPB 256
#define E_TILE 64           // C columns per WG
#define C_ELEMS (H * E_TILE)  // 32768 floats = 128KB

typedef float v2f __attribute__((ext_vector_type(2)));
typedef float v8f __attribute__((ext_vector_type(8)));
typedef int   v4i __attribute__((ext_vector_type(4)));

#if __has_builtin(__builtin_amdgcn_global_load_async_to_lds_b128) && \
    __has_builtin(__builtin_amdgcn_s_wait_asynccnt)
#define HAVE_ASYNC_LDS 1
#else
#define HAVE_ASYNC_LDS 0
#endif

__device__ __forceinline__ void g2l_b128(const float* __restrict__ g,
                                         float* __restrict__ l) {
#if HAVE_ASYNC_LDS
    auto gp = (__attribute__((address_space(1))) v4i*)((v4i*)g);
    auto lp = (__attribute__((address_space(3))) v4i*)((v4i*)l);
    __builtin_amdgcn_global_load_async_to_lds_b128(gp, lp, 0, 0);
#else
    *(float4*)l = *(const float4*)g;
#endif
}

__device__ __forceinline__ void async_stage_fence() {
#if HAVE_ASYNC_LDS
    __builtin_amdgcn_s_wait_asynccnt(0);
#endif
}

__device__ __forceinline__ void grid_sync(unsigned* bar, unsigned& gen) {
    __threadfence();
    __syncthreads();
    if (threadIdx.x == 0) {
        unsigned target = gen + 1u;
        unsigned prev = atomicAdd(&bar[0], 1u);
        if (prev == target * (unsigned)NWG - 1u) {
            atomicExch(&bar[1], target);   // release the generation
        } else {
            while (atomicAdd(&bar[1], 0u) < target) {
                __builtin_amdgcn_s_sleep(2);
            }
        }
        gen = target;
    }
    __syncthreads();
}

// fp32 WMMA 16x16 tile accumulated over K=512.
// A rows come from LDS (a_lds, row stride H; rows >= 4 are zero-padded via
// clamped address + register select, no exec-masked loads).
// B[k][n] = brow[k] with brow already offset to weight row (c0+n).
__device__ __forceinline__ v8f wmma_tile_k512(v8f acc,
                                              const float* __restrict__ a_lds,
                                              const float* __restrict__ brow,
                                              int lane) {
    const int n   = lane & 15;
    const int kh  = (lane >> 4) << 1;
    const bool ok = (n < B_BATCH);
    const float* arow = a_lds + (ok ? n : (B_BATCH - 1)) * H;
    for (int kb = 0; kb < H; kb += 128) {
        if (kb + 128 < H) __builtin_prefetch(brow + kb + 128, 0, 3);
        #pragma unroll 8
        for (int k0 = kb; k0 < kb + 128; k0 += 4) {
            float2 av = *(const float2*)(arow + k0 + kh);
            float2 bv = *(const float2*)(brow + k0 + kh);
            v2f a, b;
            a[0] = ok ? av.x : 0.0f;
            a[1] = ok ? av.y : 0.0f;
            b[0] = bv.x;
            b[1] = bv.y;
            acc = __builtin_amdgcn_wmma_f32_16x16x4_f32(false, a, false, b,
                                                        (short)0, acc, false, false);
        }
    }
    return acc;
}

__global__ void __launch_bounds__(TPB, 1)
xlstm_block_kernel(const float* __restrict__ x,
                   const float* __restrict__ Wih, const float* __restrict__ Whh,
                   const float* __restrict__ bvec,
                   const float* __restrict__ Wq, const float* __restrict__ bq,
                   const float* __restrict__ Wk, const float* __restrict__ bk,
                   const float* __restrict__ Wv, const float* __restrict__ bv,
                   const float* __restrict__ ln_g, const float* __restrict__ ln_b,
                   const float* __restrict__ Wp, const float* __restrict__ bp,
                   float* __restrict__ out,
                   unsigned* __restrict__ bar,
                   float* __restrict__ gbuf,   // [B][GCOLS]
                   float* __restrict__ hbuf,   // [B][H]
                   float* __restrict__ ys0,    // [B][T][H]
                   float* __restrict__ ys1)    // [B][T][H]
{
    __shared__ __align__(16) float smem[39040];   // 152.5 KB of the 320KB/WGP

    const int tid  = threadIdx.x;
    const int lane = tid & 31;
    const int wvid = (blockIdx.x << 3) + (tid >> 5);   // global wave id 0..255

    // phase-2 ownership: batch b, C-column slice [e0, e0+64)
    const int b  = blockIdx.x >> 3;
    const int e0 = (blockIdx.x & 7) << 6;

    float* Cs  = smem;                // [H][E_TILE]         32768
    float* sf  = smem + C_ELEMS;      // exp(f) [H]
    float* siv = sf + H;              // exp(i)*v [H]
    float* sq  = siv + H;             // q [H]
    float* sk  = sq + H;              // k slice [E_TILE]
    float* so  = sk + E_TILE;         // sigmoid(o) [E_TILE]
    float* shp = so + E_TILE;         // h partials [8][E_TILE] = 512
    float* sx  = shp + 8 * E_TILE;    // staged x_t [4][H]    2048
    float* sh  = sx + B_BATCH * H;    // staged h_prev [4][H] 2048

    unsigned gen = 0;

    for (int l = 0; l < L_LAYERS; ++l) {
        const float* Wih_l = Wih + l * THREEH * H;
        const float* Whh_l = Whh + l * THREEH * H;
        const float* b_l   = bvec + l * THREEH;
        const float* Wq_l  = Wq + l * H * H;
        const float* Wk_l  = Wk + l * H * H;
        const float* Wv_l  = Wv + l * H * H;
        const float* bq_l  = bq + l * H;
        const float* bk_l  = bk + l * H;
        const float* bv_l  = bv + l * H;
        const float* xbase = (l == 0) ? x : ys0;
        float* ysL         = (l == 0) ? ys0 : ys1;

        // reset this WG's C slice
        for (int i = tid; i < C_ELEMS; i += TPB) Cs[i] = 0.0f;
        __syncthreads();

        for (int t = 0; t < T_STEPS; ++t) {
            // ---- stage A operands (x_t rows, h_prev rows) into LDS --------
            for (int i = tid; i < 512; i += TPB) {       // 512 x 16B chunks
                int bb = i >> 7;
                int c4 = (i & 127) << 2;
                g2l_b128(xbase + (bb * T_STEPS + t) * H + c4, sx + bb * H + c4);
            }
            if (t > 0) {
                for (int i = tid; i < 512; i += TPB) {
                    int bb = i >> 7;
                    int c4 = (i & 127) << 2;
                    g2l_b128(hbuf + bb * H + c4, sh + bb * H + c4);
                }
            }
            async_stage_fence();
            __syncthreads();

            // ---------------- phase 1: [4 x 3072] gate/qkv GEMM via WMMA ----
            if (wvid < GCOLS / 16) {
                const int c0 = wvid * 16;
                const float* Wx;
                const float* Wh = nullptr;
                const float* bias_p;
                if (c0 < THREEH) {
                    Wx = Wih_l + c0 * H;
                    Wh = Whh_l + c0 * H;
                    bias_p = b_l + c0;
                } else {
                    int c1 = c0 - THREEH;
                    int sel = c1 >> 9;
                    int cc  = c1 & (H - 1);
                    Wx = (sel == 0 ? Wq_l : (sel == 1 ? Wk_l : Wv_l)) + cc * H;
                    bias_p = (sel == 0 ? bq_l : (sel == 1 ? bk_l : bv_l)) + cc;
                }
                const int n = lane & 15;
                v8f acc = {};
                acc = wmma_tile_k512(acc, sx, Wx + n * H, lane);
                if (Wh && t > 0) {
                    acc = wmma_tile_k512(acc, sh, Wh + n * H, lane);
                }
                if (lane < 16) {
                    float bval = bias_p[n];
                    #pragma unroll
                    for (int r = 0; r < B_BATCH; ++r)
                        gbuf[r * GCOLS + c0 + n] = acc[r] + bval;
                }
            }
            grid_sync(bar, gen);

            // ---------------- phase 2: gate transforms + C update + h -------
            const float* gb = gbuf + b * GCOLS;
            for (int d = tid; d < H; d += TPB) {
                sf[d]  = __expf(gb[512 + d]);                 // f gate
                siv[d] = __expf(gb[d]) * gb[2560 + d];        // i gate * v
                sq[d]  = gb[1536 + d];                        // q
            }
            if (tid < E_TILE) {
                sk[tid] = gb[2048 + e0 + tid];                // k slice
                so[tid] = 1.0f / (1.0f + __expf(-gb[1024 + e0 + tid]));
            }
            __syncthreads();
            {
                const int e2 = (tid & 31) << 1;   // e-column pair
                const int d0 = tid >> 5;          // 0..7
                const float2 kk = *(const float2*)(sk + e2);
                float accx = 0.0f, accy = 0.0f;
                #pragma unroll 4
                for (int d = d0; d < H; d += 8) {
                    float2 c = *(float2*)(Cs + d * E_TILE + e2);
                    float fd = sf[d], ivd = siv[d], qd = sq[d];
                    c.x = fd * c.x + ivd * kk.x;
                    c.y = fd * c.y + ivd * kk.y;
                    *(float2*)(Cs + d * E_TILE + e2) = c;
                    accx += qd * c.x;
                    accy += qd * c.y;
                }
                shp[d0 * E_TILE + e2]     = accx;
                shp[d0 * E_TILE + e2 + 1] = accy;
            }
            __syncthreads();
            if (tid < E_TILE) {
                float s = 0.0f;
                #pragma unroll
                for (int r = 0; r < 8; ++r) s += shp[r * E_TILE + tid];
                float hv = so[tid] * s;
                hbuf[b * H + e0 + tid] = hv;
                ysL[(b * T_STEPS + t) * H + e0 + tid] = hv;
            }
            grid_sync(bar, gen);
        }
    }

    // ---------------- epilogue: GELU -> LayerNorm -> WMMA proj + residual ---
    const int r0 = blockIdx.x * 16;     // 32 WGs x 16 rows = 512 rows (b*T+t)
    float* yn   = smem;                 // [16][H]
    float* red1 = smem + 16 * H;        // [256]
    float* red2 = red1 + TPB;           // [256]
    float* stat = red2 + TPB;           // [16][2] mu, rstd

    const int mrow = tid >> 4;          // 0..15 : row within this WG's block
    const int l16  = tid & 15;
    {
        float s = 0.0f, s2 = 0.0f;
        const float* yr = ys1 + (r0 + mrow) * H;
        for (int j = 0; j < H / 16; ++j) {
            int e = l16 + (j << 4);
            float v = yr[e];
            float g = 0.5f * v * (1.0f + erff(v * 0.70710678118654752f));
            yn[mrow * H + e] = g;
            s += g;
            s2 += g * g;
        }
        red1[tid] = s;
        red2[tid] = s2;
    }
    __syncthreads();
    for (int off = 8; off >= 1; off >>= 1) {
        if (l16 < off) {
            red1[tid] += red1[tid + off];
            red2[tid] += red2[tid + off];
        }
        __syncthreads();
    }
    if (l16 == 0) {
        float mu  = red1[tid] * (1.0f / (float)H);
        float var = red2[tid] * (1.0f / (float)H) - mu * mu;
        stat[mrow * 2]     = mu;
        stat[mrow * 2 + 1] = rsqrtf(var + 1e-5f);
    }
    __syncthreads();
    for (int idx = tid; idx < 16 * H; idx += TPB) {
        int m = idx >> 9, e = idx & (H - 1);
        yn[idx] = (yn[idx] - stat[m * 2]) * stat[m * 2 + 1] * ln_g[e] + ln_b[e];
    }
    __syncthreads();

    // out[16 x 512] = yn[16 x 512] @ Wp^T  (+ bp + residual x), WMMA tiles
    const int wv = tid >> 5;            // wave in WG: 0..7, 4 col tiles each
    const int n  = lane & 15;
    const int kh = (lane >> 4) << 1;
    const int Mb = (lane >> 4) << 3;
    for (int it = 0; it < 4; ++it) {
        const int c0 = (wv * 4 + it) * 16;
        const float* brow = Wp + (c0 + n) * H;
        const float* arow = yn + n * H;
        v8f acc = {};
        for (int kb = 0; kb < H; kb += 128) {
            if (kb + 128 < H) __builtin_prefetch(brow + kb + 128, 0, 3);
            #pragma unroll 8
            for (int k0 = kb; k0 < kb + 128; k0 += 4) {
                float2 av = *(const float2*)(arow + k0 + kh);
                float2 bb = *(const float2*)(brow + k0 + kh);
                v2f a, bw;
                a[0] = av.x;
                a[1] = av.y;
                bw[0] = bb.x;
                bw[1] = bb.y;
                acc = __builtin_amdgcn_wmma_f32_16x16x4_f32(false, a, false, bw,
                                                            (short)0, acc, false, false);
            }
        }
        const float bpn = bp[c0 + n];
        #pragma unroll
        for (int r = 0; r < 8; ++r) {
            int row = r0 + Mb + r;
            int col = c0 + n;
            out[row * IN_DIM + col] = acc[r] + bpn + x[row * IN_DIM + col];
        }
    }
}

extern "C" void kernel_launch(void* const* d_in, const int* in_sizes, int n_in,
                              void* d_out, int out_size, void* d_ws, size_t ws_size,
                              hipStream_t stream) {
    (void)in_sizes; (void)n_in; (void)out_size; (void)ws_size;
    const float* x    = (const float*)d_in[0];
    const float* Wih  = (const float*)d_in[1];
    const float* Whh  = (const float*)d_in[2];
    const float* bvec = (const float*)d_in[3];
    const float* Wq   = (const float*)d_in[4];
    const float* bq   = (const float*)d_in[5];
    const float* Wk   = (const float*)d_in[6];
    const float* bk   = (const float*)d_in[7];
    const float* Wv   = (const float*)d_in[8];
    const float* bv   = (const float*)d_in[9];
    const float* ln_g = (const float*)d_in[10];
    const float* ln_b = (const float*)d_in[11];
    const float* Wp   = (const float*)d_in[12];
    const float* bp   = (const float*)d_in[13];
    float* out = (float*)d_out;

    unsigned* bar = (unsigned*)d_ws;
    float* wsf  = (float*)((char*)d_ws + 256);
    float* gbuf = wsf;                       // B*GCOLS      = 12288 floats
    float* hbuf = gbuf + B_BATCH * GCOLS;    // B*H          = 2048
    float* ys0  = hbuf + B_BATCH * H;        // B*T*H        = 262144
    float* ys1  = ys0 + B_BATCH * T_STEPS * H;

    // barrier state must start at zero every call (graph replays re-run this)
    (void)hipMemsetAsync(bar, 0, 256, stream);

    xlstm_block_kernel<<<dim3(NWG), dim3(TPB), 0, stream>>>(
        x, Wih, Whh, bvec, Wq, bq, Wk, bk, Wv, bv, ln_g, ln_b, Wp, bp,
        out, bar, gbuf, hbuf, ys0, ys1);
}